// Quantizer_38628935860849
// MI455X (gfx1250) — compile-verified
//
#include <hip/hip_runtime.h>

typedef __attribute__((ext_vector_type(2))) float v2f;
typedef __attribute__((ext_vector_type(4))) float v4f;
typedef __attribute__((ext_vector_type(8))) float v8f;

#define DIM        64
#define KCODES     512
#define ETPAD      68           // ET row stride (floats); 68 % 64 == 4 -> lane halves on disjoint banks
#define TPB        256          // 8 waves
#define ROWS_PER_WAVE 32        // two 16-row A subtiles share each B fragment
#define ROWS_PER_BLOCK 256
#define NROWS      262144
#define NBLOCKS    (NROWS / ROWS_PER_BLOCK)   // 1024
#define NELEM      16777216
#define ETG_FLOATS (KCODES * ETPAD)           // 34816 floats in d_ws
#define SMEM_FLOATS (KCODES * ETPAD + KCODES + 8)

__device__ __forceinline__ void async_copy_b128(unsigned lds_off, const void* gsrc) {
    // GLOBAL_LOAD_ASYNC_TO_LDS_B128: per-lane 16B DMA, tracked by ASYNCcnt
    asm volatile("global_load_async_to_lds_b128 %0, %1, off"
                 :: "v"(lds_off), "v"(gsrc) : "memory");
}

// One-time: transpose + pad codebook E[64][512] -> ETg[512][68] in global scratch.
__global__ __launch_bounds__(TPB) void vq_transpose(const float* __restrict__ E,
                                                    float* __restrict__ ETg) {
    for (int i = threadIdx.x; i < DIM * KCODES; i += TPB) {
        const int r = i >> 9;        // d index
        const int c = i & 511;       // code index
        ETg[c * ETPAD + r] = E[i];   // coalesced read, scattered write (one-time, 128 KB)
    }
}

__global__ __launch_bounds__(TPB) void vq_main(const float* __restrict__ X,
                                               const float* __restrict__ ETg,
                                               float* __restrict__ out,
                                               float* __restrict__ partial) {
    extern __shared__ float smem[];
    float* ET   = smem;                  // [KCODES][ETPAD]  ET[c][d] = E[d][c]
    float* e2   = smem + KCODES * ETPAD; // [KCODES]
    float* wsum = e2 + KCODES;           // [8]

    const int tid  = threadIdx.x;
    const int lane = tid & 31;
    const int wave = tid >> 5;
    const int n    = lane & 15;
    const int half = lane >> 4;

    // ---- async DMA stage: pre-transposed image, contiguous b128 chunks ----
    {
        const unsigned etbase = (unsigned)(uintptr_t)ET;   // low 32 bits of shared addr = LDS offset
        #pragma unroll 2
        for (int i = tid; i < ETG_FLOATS / 4; i += TPB)    // 8704 chunks / 256 = 34 per thread
            async_copy_b128(etbase + (unsigned)i * 16u, ETg + i * 4);
    }

    const int rowBase = blockIdx.x * ROWS_PER_BLOCK + wave * ROWS_PER_WAVE;

    // ---- A fragments (overlap with codebook DMA): two 16x64 subtiles ----
    // ISA 16x4 fp32 A layout: lane holds M = lane&15; VGPR0 -> K = 4*kt + 2*half, VGPR1 -> +1
    const float* xrow0 = X + (size_t)(rowBase + n) * DIM + 2 * half;
    const float* xrow1 = xrow0 + 16 * DIM;
    v2f a0[16], a1[16];
    #pragma unroll
    for (int kt = 0; kt < 16; ++kt) {
        a0[kt] = *(const v2f*)(xrow0 + 4 * kt);
        a1[kt] = *(const v2f*)(xrow1 + 4 * kt);
    }

    asm volatile("s_wait_asynccnt 0" ::: "memory");
    __syncthreads();

    // ---- per-code squared norms ||e_c||^2 (rows of ET are contiguous) ----
    for (int c = tid; c < KCODES; c += TPB) {
        const float* er = ET + c * ETPAD;
        float s = 0.f;
        #pragma unroll 8
        for (int d = 0; d < DIM; ++d) s = __builtin_fmaf(er[d], er[d], s);
        e2[c] = s;
    }
    __syncthreads();

    // ---- fused GEMM + running argmin over 32 code tiles ----
    // Within a lane the candidate code is strictly increasing in t, so strict '<'
    // keeps the lowest-index minimum exactly (argmin-first semantics, no int compare).
    float bestv[16];
    int   besti[16];
    #pragma unroll
    for (int v = 0; v < 16; ++v) { bestv[v] = 3.0e38f; besti[v] = 0; }

    for (int t = 0; t < 32; ++t) {
        const float* bbase = ET + (t * 16 + n) * ETPAD + 2 * half;  // ds offsets: 16*kt
        v8f acc0 = {}, acc1 = {};
        #pragma unroll
        for (int kt = 0; kt < 16; ++kt) {
            v2f b = *(const v2f*)(bbase + 4 * kt);                  // ds_load_2addr_b64 pairs
            acc0 = __builtin_amdgcn_wmma_f32_16x16x4_f32(
                false, a0[kt], false, b, (short)0, acc0, false, false);
            acc1 = __builtin_amdgcn_wmma_f32_16x16x4_f32(
                false, a1[kt], false, b, (short)0, acc1, false, false);
        }
        const int   code = t * 16 + n;
        const float e2v  = e2[code];
        #pragma unroll
        for (int v = 0; v < 8; ++v) {
            // ||x||^2 is row-constant -> irrelevant for argmin; score = ||e||^2 - 2*sim
            float s0 = __builtin_fmaf(-2.f, acc0[v], e2v);
            bool  c0 = s0 < bestv[v];
            bestv[v] = c0 ? s0 : bestv[v];
            besti[v] = c0 ? code : besti[v];
            float s1 = __builtin_fmaf(-2.f, acc1[v], e2v);
            bool  c1 = s1 < bestv[8 + v];
            bestv[8 + v] = c1 ? s1 : bestv[8 + v];
            besti[8 + v] = c1 ? code : besti[8 + v];
        }
    }

    // ---- argmin reduction across the 16 lanes of each half (tie-break: lowest index) ----
    #pragma unroll
    for (int v = 0; v < 16; ++v) {
        float bv = bestv[v];
        int   bi = besti[v];
        #pragma unroll
        for (int m = 1; m < 16; m <<= 1) {
            float ov = __shfl_xor(bv, m, 32);
            int   oi = __shfl_xor(bi, m, 32);
            if (ov < bv || (ov == bv && oi < bi)) { bv = ov; bi = oi; }
        }
        besti[v] = bi;                  // uniform within each half
    }

    // ---- gather code rows (contiguous in ET), write quantized, accumulate SSE ----
    float lossAcc = 0.f;
    #pragma unroll
    for (int v = 0; v < 16; ++v) {
        const int sub = v >> 3;
        const int row = rowBase + sub * 16 + (v & 7) + half * 8;
        const int idx = besti[v];
        v4f q  = *(const v4f*)(ET + idx * ETPAD + 4 * n);          // ds_load_b128, full-bank
        v4f xv = *(const v4f*)(X + (size_t)row * DIM + 4 * n);     // global_load_b128
        v4f df = q - xv;
        lossAcc = __builtin_fmaf(df.x, df.x, lossAcc);
        lossAcc = __builtin_fmaf(df.y, df.y, lossAcc);
        lossAcc = __builtin_fmaf(df.z, df.z, lossAcc);
        lossAcc = __builtin_fmaf(df.w, df.w, lossAcc);
        *(v4f*)(out + (size_t)row * DIM + 4 * n) = q;              // global_store_b128
    }

    // ---- deterministic block reduction of SSE ----
    #pragma unroll
    for (int m = 1; m < 32; m <<= 1)
        lossAcc += __shfl_xor(lossAcc, m, 32);
    if (lane == 0) wsum[wave] = lossAcc;
    __syncthreads();
    if (tid == 0) {
        float s = 0.f;
        #pragma unroll
        for (int w = 0; w < 8; ++w) s += wsum[w];
        partial[blockIdx.x] = s;
    }
}

__global__ void vq_reduce(const float* __restrict__ partial, float* __restrict__ out) {
    if (threadIdx.x == 0 && blockIdx.x == 0) {
        float s = 0.f;
        for (int i = 0; i < NBLOCKS; ++i) s += partial[i];   // fixed order -> deterministic
        const float mean = s * (1.0f / (float)NELEM);
        out[NELEM]     = 0.25f * mean;   // commitment_loss (BETA = 0.25)
        out[NELEM + 1] = mean;           // codebook_loss
    }
}

extern "C" void kernel_launch(void* const* d_in, const int* in_sizes, int n_in,
                              void* d_out, int out_size, void* d_ws, size_t ws_size,
                              hipStream_t stream) {
    (void)in_sizes; (void)n_in; (void)out_size; (void)ws_size;
    const float* X = (const float*)d_in[0];   // [64,64,64,64] fp32
    const float* E = (const float*)d_in[1];   // [64,512] fp32
    float* out     = (float*)d_out;           // 16777216 quantized + 2 scalars
    float* ETg     = (float*)d_ws;            // [512][68] transposed codebook
    float* partial = ETg + ETG_FLOATS;        // [NBLOCKS]

    vq_transpose<<<1, TPB, 0, stream>>>(E, ETg);
    vq_main<<<NBLOCKS, TPB, SMEM_FLOATS * sizeof(float), stream>>>(X, ETg, out, partial);
    vq_reduce<<<1, 32, 0, stream>>>(partial, out);
}